// GCNClassifier_64579128263107
// MI455X (gfx1250) — compile-verified
//
#include <hip/hip_runtime.h>
#include <hip/hip_bf16.h>

typedef __attribute__((ext_vector_type(2))) float v2f;
typedef __attribute__((ext_vector_type(8))) float v8f;

#define HID 64

// ---------------- degree / norm ----------------
__global__ void degree_kernel(const int* __restrict__ dst, float* __restrict__ deg, int E) {
    int tid = blockIdx.x * blockDim.x + threadIdx.x;
    if (tid >= E) return;
    atomicAdd(&deg[dst[tid]], 1.0f);
}

__global__ void dinv_kernel(const float* __restrict__ deg, float* __restrict__ dinv, int N) {
    int tid = blockIdx.x * blockDim.x + threadIdx.x;
    if (tid >= N) return;
    dinv[tid] = rsqrtf(deg[tid] + 1.0f);   // +1 for self loop; deg >= 1 always
}

// ---------------- layer-1 transform: h = x @ W1  (x: [N,2], W1: [2,64]) ----------------
__global__ void transform1_kernel(const float* __restrict__ x, const float* __restrict__ W1,
                                  float* __restrict__ h, int N) {
    long tid = (long)blockIdx.x * blockDim.x + threadIdx.x;
    if (tid >= (long)N * HID) return;
    int n = (int)(tid >> 6), j = (int)(tid & 63);
    h[tid] = x[n * 2 + 0] * W1[j] + x[n * 2 + 1] * W1[HID + j];
}

// ---------------- edge scatter: agg[dst] += h[src] * dinv[src]*dinv[dst] ----------------
// One wave per edge; each of 32 lanes handles 2 of the 64 feature dims (float2 load + 2 f32 atomics).
// src/dst/dinv reads are wave-uniform -> coalesced to single requests.
__global__ void scatter_kernel(const float* __restrict__ h, const int* __restrict__ src,
                               const int* __restrict__ dst, const float* __restrict__ dinv,
                               float* __restrict__ agg, int E) {
    long tid = (long)blockIdx.x * blockDim.x + threadIdx.x;
    int e = (int)(tid >> 5);
    if (e >= E) return;
    int lane = (int)(tid & 31);
    int s = src[e], d = dst[e];
    float norm = dinv[s] * dinv[d];
    const float2* hp = (const float2*)(h + (long)s * HID);
    float2 v = hp[lane];
    float* ap = agg + (long)d * HID + lane * 2;
    atomicAdd(ap + 0, v.x * norm);
    atomicAdd(ap + 1, v.y * norm);
}

// ---------------- self-loop + bias + relu (in place into agg) ----------------
__global__ void combine_kernel(const float* __restrict__ h, float* __restrict__ agg,
                               const float* __restrict__ dinv, const float* __restrict__ bias,
                               int N) {
    long tid = (long)blockIdx.x * blockDim.x + threadIdx.x;
    if (tid >= (long)N * HID) return;
    int n = (int)(tid >> 6), j = (int)(tid & 63);
    float di = dinv[n];
    float v = agg[tid] + h[tid] * di * di + bias[j];
    agg[tid] = fmaxf(v, 0.0f);
}

// ---------------- fp32 WMMA GEMM: C[M,64] = A[M,64] @ B[64,64] (+bias, relu) ----------------
// One wave computes a 16x16 output tile via 16x V_WMMA_F32_16X16X4_F32 (K=64).
__global__ void wmma_gemm64_kernel(const float* __restrict__ A, const float* __restrict__ B,
                                   const float* __restrict__ bias, float* __restrict__ C,
                                   int M, int fuse_bias_relu) {
    int wave = (int)((blockIdx.x * blockDim.x + threadIdx.x) >> 5);
    int lane = threadIdx.x & 31;
    int mt = wave >> 2;          // 64/16 = 4 N-tiles
    int nt = wave & 3;
    if (mt * 16 >= M) return;    // wave-uniform: EXEC stays all-ones inside live waves
    int m0 = mt * 16, n0 = nt * 16;
    int row  = lane & 15;        // M index for A, N index for B/C
    int half = lane >> 4;

    v8f acc = {};
#pragma unroll
    for (int k0 = 0; k0 < HID; k0 += 4) {
        int ka = k0 + half * 2;
        v2f a, b;
        a.x = A[(long)(m0 + row) * HID + ka + 0];
        a.y = A[(long)(m0 + row) * HID + ka + 1];
        b.x = B[(long)(ka + 0) * HID + n0 + row];
        b.y = B[(long)(ka + 1) * HID + n0 + row];
        acc = __builtin_amdgcn_wmma_f32_16x16x4_f32(
            /*neg_a=*/false, a, /*neg_b=*/false, b,
            /*c_mod=*/(short)0, acc, /*reuse_a=*/false, /*reuse_b=*/false);
    }

#pragma unroll
    for (int r = 0; r < 8; ++r) {
        int mm = m0 + r + half * 8;
        int nn = n0 + row;
        float v = acc[r];
        if (fuse_bias_relu) { v += bias[nn]; v = fmaxf(v, 0.0f); }
        C[(long)mm * HID + nn] = v;
    }
}

// ---------------- global mean pool ----------------
__global__ void pool_acc_kernel(const float* __restrict__ h, const int* __restrict__ batch,
                                float* __restrict__ sums, float* __restrict__ cnts, int N) {
    long tid = (long)blockIdx.x * blockDim.x + threadIdx.x;
    if (tid >= (long)N * HID) return;
    int n = (int)(tid >> 6), j = (int)(tid & 63);
    int g = batch[n];
    atomicAdd(&sums[(long)g * HID + j], h[tid]);
    if (j == 0) atomicAdd(&cnts[g], 1.0f);
}

__global__ void pool_mean_kernel(const float* __restrict__ sums, const float* __restrict__ cnts,
                                 float* __restrict__ g, int G) {
    int tid = blockIdx.x * blockDim.x + threadIdx.x;
    if (tid >= G * HID) return;
    int gi = tid >> 6;
    g[tid] = sums[tid] / fmaxf(cnts[gi], 1.0f);
}

// ---------------- final tiny FC: out[G,4] = g1 @ Wf2 + bf2 ----------------
__global__ void fc2_kernel(const float* __restrict__ g1, const float* __restrict__ Wf2,
                           const float* __restrict__ bf2, float* __restrict__ out, int G) {
    int tid = blockIdx.x * blockDim.x + threadIdx.x;
    if (tid >= G * 4) return;
    int gi = tid >> 2, o = tid & 3;
    float s = bf2[o];
#pragma unroll 8
    for (int k = 0; k < HID; ++k) s += g1[gi * HID + k] * Wf2[k * 4 + o];
    out[tid] = s;
}

extern "C" void kernel_launch(void* const* d_in, const int* in_sizes, int n_in,
                              void* d_out, int out_size, void* d_ws, size_t ws_size,
                              hipStream_t stream) {
    (void)n_in; (void)ws_size;
    const float* x     = (const float*)d_in[0];
    const int*   ei    = (const int*)  d_in[1];
    const int*   batch = (const int*)  d_in[2];
    const float* W1    = (const float*)d_in[4];
    const float* b1    = (const float*)d_in[5];
    const float* W2    = (const float*)d_in[6];
    const float* b2    = (const float*)d_in[7];
    const float* Wf1   = (const float*)d_in[8];
    const float* bf1   = (const float*)d_in[9];
    const float* Wf2   = (const float*)d_in[10];
    const float* bf2   = (const float*)d_in[11];
    float* out = (float*)d_out;

    int N = in_sizes[0] / 2;
    int E = in_sizes[1] / 2;
    int G = out_size / 4;
    const int* src = ei;
    const int* dst = ei + E;

    // workspace carve-up (floats)
    float* ws = (float*)d_ws;
    size_t off = 0;
    float* deg  = ws + off; off += (size_t)N;
    float* dinv = ws + off; off += (size_t)N;
    float* bufA = ws + off; off += (size_t)N * HID;   // transformed features
    float* bufB = ws + off; off += (size_t)N * HID;   // agg -> activations (in place)
    float* sums = ws + off; off += (size_t)G * HID;
    float* cnts = ws + off; off += (size_t)G;
    float* gbuf = ws + off; off += (size_t)G * HID;
    float* g1   = ws + off; off += (size_t)G * HID;

    const int TB = 256;
    long NH = (long)N * HID;
    long EW = (long)E * 32;

    // degree + symmetric norm
    hipMemsetAsync(deg, 0, (size_t)N * sizeof(float), stream);
    degree_kernel<<<(E + TB - 1) / TB, TB, 0, stream>>>(dst, deg, E);
    dinv_kernel<<<(N + TB - 1) / TB, TB, 0, stream>>>(deg, dinv, N);

    // ---- GCN layer 1 ----
    transform1_kernel<<<(int)((NH + TB - 1) / TB), TB, 0, stream>>>(x, W1, bufA, N);
    hipMemsetAsync(bufB, 0, (size_t)NH * sizeof(float), stream);
    scatter_kernel<<<(int)((EW + TB - 1) / TB), TB, 0, stream>>>(bufA, src, dst, dinv, bufB, E);
    combine_kernel<<<(int)((NH + TB - 1) / TB), TB, 0, stream>>>(bufA, bufB, dinv, b1, N);  // bufB = h1

    // ---- GCN layer 2: WMMA fp32 GEMM h1 @ W2 -> bufA ----
    {
        int tiles = ((N + 15) / 16) * 4;
        wmma_gemm64_kernel<<<(tiles + 7) / 8, TB, 0, stream>>>(bufB, W2, nullptr, bufA, N, 0);
    }
    hipMemsetAsync(bufB, 0, (size_t)NH * sizeof(float), stream);
    scatter_kernel<<<(int)((EW + TB - 1) / TB), TB, 0, stream>>>(bufA, src, dst, dinv, bufB, E);
    combine_kernel<<<(int)((NH + TB - 1) / TB), TB, 0, stream>>>(bufA, bufB, dinv, b2, N);  // bufB = h2

    // ---- mean pool ----
    hipMemsetAsync(sums, 0, (size_t)(G * HID + G) * sizeof(float), stream);  // sums + cnts contiguous
    pool_acc_kernel<<<(int)((NH + TB - 1) / TB), TB, 0, stream>>>(bufB, batch, sums, cnts, N);
    pool_mean_kernel<<<(G * HID + TB - 1) / TB, TB, 0, stream>>>(sums, cnts, gbuf, G);

    // ---- FC1 (WMMA, fused bias+relu) + FC2 ----
    {
        int tiles = ((G + 15) / 16) * 4;
        wmma_gemm64_kernel<<<(tiles + 7) / 8, TB, 0, stream>>>(gbuf, Wf1, bf1, g1, G, 1);
    }
    fc2_kernel<<<(G * 4 + TB - 1) / TB, TB, 0, stream>>>(g1, Wf2, bf2, out, G);
}